// CategoricalActivation_59691455480300
// MI455X (gfx1250) — compile-verified
//
#include <hip/hip_runtime.h>
#include <stdint.h>

// Problem constants (fixed by setup_inputs()).
#define S_DIM  2048
#define B_DIM  64
#define H_DIM  512
#define BH_DIM (B_DIM * H_DIM)   // 32768
#define NB     9                  // num_classes - 1
// JAX bool arrays are 1 byte/element; masks read as u8 (nonzero = true).

typedef uint32_t u32x4 __attribute__((ext_vector_type(4)));
typedef uint32_t u32x8 __attribute__((ext_vector_type(8)));
typedef float    f32x4 __attribute__((ext_vector_type(4)));  // native vec for NT builtins

// Fast softsign: v * v_rcp_f32(1+|v|).  rcp runs on the transcendental pipe
// (co-executes with VALU), replacing the ~10-op IEEE div expansion.
// MUST be the identical sequence in both kernels so self-comparisons tie.
__device__ __forceinline__ float softsign_fast(float v) {
  return v * __builtin_amdgcn_rcpf(1.0f + fabsf(v));
}

// ---------------------------------------------------------------------------
// Kernel A: gather + softsign the 9 per-column boundaries into workspace.
// bnd[k*BH + bh] = softsign(x[idx[k,bh]*BH + bh]).  294,912 threads total.
// ---------------------------------------------------------------------------
__global__ __launch_bounds__(256) void gather_boundaries(
    const float* __restrict__ x, const int* __restrict__ bidx,
    float* __restrict__ bnd) {
  int i = blockIdx.x * 256 + threadIdx.x;
  if (i >= NB * BH_DIM) return;
  int bh = i & (BH_DIM - 1);
  int s  = bidx[i];
  s = s < 0 ? 0 : (s >= S_DIM ? S_DIM - 1 : s);
  float v = x[(size_t)s * BH_DIM + bh];
  bnd[i] = softsign_fast(v);
}

// ---------------------------------------------------------------------------
// Kernel B: streaming elementwise pass.
// Block = 256 threads (8 wave32), owns 1024 consecutive columns x 32 rows.
// Boundary tile [9 x 1024] staged into LDS with one TDM tensor_load_to_lds.
// ---------------------------------------------------------------------------
__global__ __launch_bounds__(256) void cat_act_main(
    const float* __restrict__ x,
    const unsigned char* __restrict__ catm,
    const unsigned char* __restrict__ ordm,
    const int* __restrict__ perm,
    const float* __restrict__ bnd,
    float* __restrict__ out) {
  constexpr int COLS = 1024;  // columns per block
  constexpr int ROWS = 32;    // rows per block (2048 / gridDim.y)

  __shared__ float sB[NB * COLS];  // 36 KB boundary tile
  __shared__ float sT[16];         // perm LUT: T[c] = (c>=5) ? perm[c-5] : 0

  const int tid = threadIdx.x;
  const int bh0 = blockIdx.x * COLS;

  // --- Wave 0: one TDM descriptor copies the whole [9 x 1024] tile -------
  if (tid < 32) {  // exactly wave 0 in wave32; EXEC all-ones within it
    uint64_t g   = (uint64_t)(uintptr_t)(bnd + bh0);  // tile start (bytes)
    uint32_t lds = (uint32_t)(uintptr_t)(&sB[0]);     // low 32 bits = LDS byte addr

    u32x4 d0;                       // D# group 0 (128b)
    d0[0] = 1u;                     // count=1, is_restore=0, gather off
    d0[1] = lds;                    // lds_addr
    d0[2] = (uint32_t)g;            // global_addr[31:0]
    d0[3] = ((uint32_t)(g >> 32) & 0x01FFFFFFu) | (2u << 30);  // addr[56:32] | type=2

    u32x8 d1;                       // D# group 1 (256b)
    d1[0] = 2u << 16;               // workgroup_mask=0 (not in cluster), data_size=4B
    d1[1] = ((uint32_t)COLS & 0xFFFFu) << 16;              // tensor_dim0[15:0]
    d1[2] = ((uint32_t)COLS >> 16) | ((uint32_t)NB << 16); // dim0[31:16] | dim1[15:0]
    d1[3] = ((uint32_t)COLS & 0xFFFFu) << 16;              // dim1 hi=0 | tile_dim0=1024
    d1[4] = (uint32_t)NB;                                  // tile_dim1=9 | tile_dim2=0
    d1[5] = (uint32_t)BH_DIM;                              // tensor_dim0_stride=32768
    d1[6] = 0u;                                            // stride0 hi | stride1 lo
    d1[7] = 0u;

    // 2-group form (VADDR2/3 = NULL): tensors up to 2D.
    asm volatile("tensor_load_to_lds %0, %1" :: "s"(d0), "s"(d1) : "memory");
    __builtin_amdgcn_s_wait_tensorcnt(0);
  }
  // perm LUT: ord&cat positions hold counts-5 (exact int in [-5,4]);
  // exact && in-range  <=>  counts>=5, so T[c] = (c>=5) ? perm[c-5] : 0.
  if (tid < 10) sT[tid] = (tid >= 5) ? (float)perm[tid - 5] : 0.0f;
  __syncthreads();

  // --- Hoist this thread's 4 columns of boundaries into registers --------
  float bc[4][NB];
#pragma unroll
  for (int k = 0; k < NB; ++k) {
    const f32x4 t = *(const f32x4*)&sB[k * COLS + tid * 4];
    bc[0][k] = t.x; bc[1][k] = t.y; bc[2][k] = t.z; bc[3][k] = t.w;
  }
  const uint32_t cm = *(const uint32_t*)(catm + bh0 + tid * 4);
  const uint32_t om = *(const uint32_t*)(ordm + bh0 + tid * 4);
  bool cat[4], oc[4];
#pragma unroll
  for (int j = 0; j < 4; ++j) {
    bool c = ((cm >> (8 * j)) & 0xFFu) != 0u;
    bool o = ((om >> (8 * j)) & 0xFFu) != 0u;
    cat[j] = c;
    oc[j]  = c && o;  // ord_mask = ord_rand & cat_mask
  }

  // --- Stream ROWS rows: coalesced NT b128 load/store, prefetch ahead ----
  const size_t base = (size_t)blockIdx.y * ROWS * BH_DIM + bh0 + tid * 4;
  const float* xp = x + base;
  float*       op = out + base;

  for (int r = 0; r < ROWS; ++r) {
    __builtin_prefetch(xp + 4ull * BH_DIM, 0, 0);  // global_prefetch_b8
    const f32x4 xv = __builtin_nontemporal_load((const f32x4*)xp);  // th:NT b128
    float vin[4] = {xv.x, xv.y, xv.z, xv.w};
    float vout[4];
#pragma unroll
    for (int j = 0; j < 4; ++j) {
      float v = vin[j];
      float y = softsign_fast(v);
      float c = 0.0f;
#pragma unroll
      for (int k = 0; k < NB; ++k) c += (y > bc[j][k]) ? 1.0f : 0.0f;
      float pv = sT[(int)c];                   // ds_load_b32 LUT (DS pipe)
      float rv = cat[j] ? (c - 5.0f) : y;      // nc/2 == 5.0
      vout[j] = oc[j] ? pv : rv;
    }
    f32x4 ov; ov.x = vout[0]; ov.y = vout[1]; ov.z = vout[2]; ov.w = vout[3];
    __builtin_nontemporal_store(ov, (f32x4*)op);  // th:NT b128 (write-once stream)
    xp += BH_DIM;
    op += BH_DIM;
  }
}

// ---------------------------------------------------------------------------
extern "C" void kernel_launch(void* const* d_in, const int* in_sizes, int n_in,
                              void* d_out, int out_size, void* d_ws, size_t ws_size,
                              hipStream_t stream) {
  (void)in_sizes; (void)n_in; (void)out_size; (void)ws_size;
  const float*         x    = (const float*)d_in[0];
  const int*           bidx = (const int*)d_in[1];
  const unsigned char* catm = (const unsigned char*)d_in[2];  // jnp.bool_ = u8
  const unsigned char* ordm = (const unsigned char*)d_in[3];  // jnp.bool_ = u8
  const int*           perm = (const int*)d_in[4];
  // d_in[5] = num_classes (fixed at 10; hardcoded)
  float* out = (float*)d_out;
  float* bnd = (float*)d_ws;  // NB * BH_DIM floats = 1.125 MiB scratch

  gather_boundaries<<<(NB * BH_DIM + 255) / 256, 256, 0, stream>>>(x, bidx, bnd);

  dim3 grid(BH_DIM / 1024, S_DIM / 32);  // 32 x 64 blocks
  cat_act_main<<<grid, 256, 0, stream>>>(x, catm, ordm, perm, bnd, out);
}